// GP_Model_Gibbs_44229573214647
// MI455X (gfx1250) — compile-verified
//
#include <hip/hip_runtime.h>
#include <math.h>

#define N 1024
#define BK 16
#define NB (N / BK)   // 64
#define ND 20
#define STEPS 10
#define G 100
#define JITTER 1e-6f
#define PI_F 3.14159265358979f

typedef __attribute__((ext_vector_type(2))) float v2f;
typedef __attribute__((ext_vector_type(8))) float v8f;

// ---------------- RNG helpers (hash-based, deterministic) ----------------
__device__ __forceinline__ unsigned hash_u32(unsigned x) {
  x ^= x >> 16; x *= 0x7feb352dU;
  x ^= x >> 15; x *= 0x846ca68bU;
  x ^= x >> 16;
  return x;
}
__device__ __forceinline__ float uniform01(unsigned h) {
  // (0,1]
  return ((h >> 8) + 1u) * (1.0f / 16777217.0f);
}
__device__ __forceinline__ float normal_from(unsigned s) {
  float u1 = uniform01(hash_u32(s * 2654435761u + 0x9e3779b9u));
  float u2 = uniform01(hash_u32(s * 2246822519u + 0x85ebca6bu));
  return sqrtf(-2.0f * logf(u1)) * cosf(2.0f * PI_F * u2);
}

// ---------------- misc small kernels ----------------
__global__ void row_norms_k(const float* __restrict__ X, float* __restrict__ x2) {
  int i = blockIdx.x * blockDim.x + threadIdx.x;
  if (i < N) {
    float s = 0.0f;
    for (int j = 0; j < 64; ++j) { float v = X[i * 64 + j]; s += v * v; }
    x2[i] = s;
  }
}

__global__ void kappa_init_k(const int* __restrict__ Y, float* __restrict__ kappa) {
  int i = blockIdx.x * blockDim.x + threadIdx.x;
  if (i < N) kappa[i] = (Y[i] != 0) ? 0.5f : -0.5f;
}

// mode 0: dst = src + scalar*I ; mode 1: dst = src + diag(vec) ; mode 2: dst = src + diag(1/clip(vec))
__global__ void form_matrix_k(float* __restrict__ dst, const float* __restrict__ src,
                              const float* __restrict__ vec, float scalar, int mode) {
  for (int idx = blockIdx.x * blockDim.x + threadIdx.x; idx < N * N;
       idx += gridDim.x * blockDim.x) {
    float v = src[idx];
    int r = idx >> 10, c = idx & (N - 1);
    if (r == c) {
      if (mode == 0) v += scalar;
      else if (mode == 1) v += vec[r];
      else { float om = vec[r]; om = om > 1e-16f ? om : 1e-16f; v += 1.0f / om; }
    }
    dst[idx] = v;
  }
}

__global__ void set_identity_k(float* __restrict__ M) {
  for (int idx = blockIdx.x * blockDim.x + threadIdx.x; idx < N * N;
       idx += gridDim.x * blockDim.x) {
    int r = idx >> 10, c = idx & (N - 1);
    M[idx] = (r == c) ? 1.0f : 0.0f;
  }
}

// ---------------- RBF kernel via FP32 WMMA (Gram = X X^T) ----------------
__global__ void rbf_wmma_k(const float* __restrict__ X, const float* __restrict__ x2,
                           const float* __restrict__ osp, const float* __restrict__ lsp,
                           float* __restrict__ K) {
  int bi = blockIdx.x, bj = blockIdx.y;
  int lane = threadIdx.x, half = lane >> 4, l16 = lane & 15;
  v8f acc = {};
  const float* arow = X + (size_t)(bi * 16 + l16) * 64;
  const float* brow = X + (size_t)(bj * 16 + l16) * 64;
  for (int k = 0; k < 64; k += 4) {
    v2f a, b;
    a.x = arow[k + 2 * half];
    a.y = arow[k + 2 * half + 1];
    b.x = brow[k + 2 * half];
    b.y = brow[k + 2 * half + 1];
    acc = __builtin_amdgcn_wmma_f32_16x16x4_f32(false, a, false, b, (short)0, acc,
                                                false, false);
  }
  float os = osp[0], ls = lsp[0];
  float inv2 = 0.5f / (ls * ls);
  for (int v = 0; v < 8; ++v) {
    int r = bi * 16 + v + 8 * half;
    int c = bj * 16 + l16;
    float d2 = x2[r] + x2[c] - 2.0f * acc[v];
    d2 = d2 > 0.0f ? d2 : 0.0f;
    K[(size_t)r * N + c] = os * expf(-d2 * inv2);
  }
}

// ---------------- blocked Cholesky (right-looking, 16x16 blocks) ----------------
__global__ void chol_diag_k(float* __restrict__ M, int jb) {
  __shared__ float a[16][17];
  int t = threadIdx.x;  // 32 threads
  for (int e = t; e < 256; e += 32)
    a[e >> 4][e & 15] = M[(size_t)(jb * 16 + (e >> 4)) * N + jb * 16 + (e & 15)];
  __syncthreads();
  for (int c = 0; c < 16; ++c) {
    if (t == 0) a[c][c] = sqrtf(fmaxf(a[c][c], 1e-30f));
    __syncthreads();
    if (t > c && t < 16) a[t][c] /= a[c][c];
    __syncthreads();
    if (t < 16 && t > c)
      for (int q = c + 1; q <= t; ++q) a[t][q] -= a[t][c] * a[q][c];
    __syncthreads();
  }
  for (int e = t; e < 256; e += 32) {
    int r = e >> 4, cc = e & 15;
    M[(size_t)(jb * 16 + r) * N + jb * 16 + cc] = (cc <= r) ? a[r][cc] : 0.0f;
  }
}

__global__ void chol_panel_k(float* __restrict__ M, int jb) {
  __shared__ float Ljj[16][17];
  int tid = threadIdx.x;  // 128
  for (int e = tid; e < 256; e += blockDim.x)
    Ljj[e >> 4][e & 15] = M[(size_t)(jb * 16 + (e >> 4)) * N + jb * 16 + (e & 15)];
  __syncthreads();
  int i = (jb + 1) * 16 + blockIdx.x * blockDim.x + tid;
  if (i >= N) return;
  float* row = M + (size_t)i * N + jb * 16;
  float x[16];
  for (int c = 0; c < 16; ++c) {
    float s = row[c];
    for (int q = 0; q < c; ++q) s -= x[q] * Ljj[c][q];
    x[c] = s / Ljj[c][c];
  }
  for (int c = 0; c < 16; ++c) row[c] = x[c];
}

// trailing update: C(ti,tj) -= P(ti) P(tj)^T, via 4x WMMA f32 16x16x4
__global__ void chol_syrk_k(float* __restrict__ M, int jb) {
  int ti = jb + 1 + blockIdx.x;
  int tj = jb + 1 + blockIdx.y;
  if (ti < tj) return;  // wave-uniform early exit; lower triangle only
  int lane = threadIdx.x, half = lane >> 4, l16 = lane & 15;
  const float* Pi = M + (size_t)(ti * 16 + l16) * N + jb * 16;
  const float* Pj = M + (size_t)(tj * 16 + l16) * N + jb * 16;
  v8f acc;
  for (int v = 0; v < 8; ++v)
    acc[v] = M[(size_t)(ti * 16 + v + 8 * half) * N + tj * 16 + l16];
  for (int kk = 0; kk < 16; kk += 4) {
    v2f a, b;
    a.x = -Pi[kk + 2 * half];
    a.y = -Pi[kk + 2 * half + 1];
    b.x = Pj[kk + 2 * half];
    b.y = Pj[kk + 2 * half + 1];
    acc = __builtin_amdgcn_wmma_f32_16x16x4_f32(false, a, false, b, (short)0, acc,
                                                false, false);
  }
  for (int v = 0; v < 8; ++v)
    M[(size_t)(ti * 16 + v + 8 * half) * N + tj * 16 + l16] = acc[v];
}

// ---------------- blocked TRSM (N RHS) for Kinv = L^-T L^-1 ----------------
// forward: Y[jb-rows, :] -= L[jb-rows, 0:jb*16] @ Y[0:jb*16, :]
__global__ void trsm_fwd_gemm_k(float* __restrict__ Ym, const float* __restrict__ L,
                                int jb) {
  int tc = blockIdx.x;
  int lane = threadIdx.x, half = lane >> 4, l16 = lane & 15;
  v8f acc;
  for (int v = 0; v < 8; ++v)
    acc[v] = Ym[(size_t)(jb * 16 + v + 8 * half) * N + tc * 16 + l16];
  const float* Arow = L + (size_t)(jb * 16 + l16) * N;
  int kmax = jb * 16;
  for (int k = 0; k < kmax; k += 4) {
    __builtin_prefetch(Arow + k + 64, 0, 1);
    v2f a, b;
    a.x = -Arow[k + 2 * half];
    a.y = -Arow[k + 2 * half + 1];
    b.x = Ym[(size_t)(k + 2 * half) * N + tc * 16 + l16];
    b.y = Ym[(size_t)(k + 2 * half + 1) * N + tc * 16 + l16];
    acc = __builtin_amdgcn_wmma_f32_16x16x4_f32(false, a, false, b, (short)0, acc,
                                                false, false);
  }
  for (int v = 0; v < 8; ++v)
    Ym[(size_t)(jb * 16 + v + 8 * half) * N + tc * 16 + l16] = acc[v];
}

__global__ void trsm_fwd_small_k(float* __restrict__ Ym, const float* __restrict__ L,
                                 int jb) {
  __shared__ float Ljj[16][17];
  int tid = threadIdx.x;  // 256
  Ljj[tid >> 4][tid & 15] = L[(size_t)(jb * 16 + (tid >> 4)) * N + jb * 16 + (tid & 15)];
  __syncthreads();
  int col = blockIdx.x * blockDim.x + tid;
  float y[16];
  for (int r = 0; r < 16; ++r) y[r] = Ym[(size_t)(jb * 16 + r) * N + col];
  for (int r = 0; r < 16; ++r) {
    float s = y[r];
    for (int q = 0; q < r; ++q) s -= Ljj[r][q] * y[q];
    y[r] = s / Ljj[r][r];
  }
  for (int r = 0; r < 16; ++r) Ym[(size_t)(jb * 16 + r) * N + col] = y[r];
}

// backward: Y[jb-rows, :] -= L[(jb+1)*16:N, jb-cols]^T @ Y[(jb+1)*16:N, :]
__global__ void trsm_bwd_gemm_k(float* __restrict__ Ym, const float* __restrict__ L,
                                int jb) {
  int tc = blockIdx.x;
  int lane = threadIdx.x, half = lane >> 4, l16 = lane & 15;
  v8f acc;
  for (int v = 0; v < 8; ++v)
    acc[v] = Ym[(size_t)(jb * 16 + v + 8 * half) * N + tc * 16 + l16];
  for (int k = (jb + 1) * 16; k < N; k += 4) {
    v2f a, b;
    a.x = -L[(size_t)(k + 2 * half) * N + jb * 16 + l16];
    a.y = -L[(size_t)(k + 2 * half + 1) * N + jb * 16 + l16];
    b.x = Ym[(size_t)(k + 2 * half) * N + tc * 16 + l16];
    b.y = Ym[(size_t)(k + 2 * half + 1) * N + tc * 16 + l16];
    acc = __builtin_amdgcn_wmma_f32_16x16x4_f32(false, a, false, b, (short)0, acc,
                                                false, false);
  }
  for (int v = 0; v < 8; ++v)
    Ym[(size_t)(jb * 16 + v + 8 * half) * N + tc * 16 + l16] = acc[v];
}

__global__ void trsm_bwd_small_k(float* __restrict__ Ym, const float* __restrict__ L,
                                 int jb) {
  __shared__ float Ljj[16][17];
  int tid = threadIdx.x;  // 256
  Ljj[tid >> 4][tid & 15] = L[(size_t)(jb * 16 + (tid >> 4)) * N + jb * 16 + (tid & 15)];
  __syncthreads();
  int col = blockIdx.x * blockDim.x + tid;
  float y[16];
  for (int r = 0; r < 16; ++r) y[r] = Ym[(size_t)(jb * 16 + r) * N + col];
  for (int r = 15; r >= 0; --r) {
    float s = y[r];
    for (int q = r + 1; q < 16; ++q) s -= Ljj[q][r] * y[q];
    y[r] = s / Ljj[r][r];
  }
  for (int r = 0; r < 16; ++r) Ym[(size_t)(jb * 16 + r) * N + col] = y[r];
}

// ---------------- vector triangular solves (single block) ----------------
__global__ void fwd_solve_vec_k(const float* __restrict__ L, const float* __restrict__ b,
                                float* __restrict__ x) {
  __shared__ float xs[N];
  __shared__ float red[256];
  int tid = threadIdx.x;
  for (int i = tid; i < N; i += 256) xs[i] = b[i];
  __syncthreads();
  for (int i = 0; i < N; ++i) {
    const float* row = L + (size_t)i * N;
    float part = 0.0f;
    for (int j = tid; j < i; j += 256) part += row[j] * xs[j];
    red[tid] = part;
    __syncthreads();
    for (int o = 128; o > 0; o >>= 1) {
      if (tid < o) red[tid] += red[tid + o];
      __syncthreads();
    }
    if (tid == 0) xs[i] = (xs[i] - red[0]) / row[i];
    __syncthreads();
  }
  for (int i = tid; i < N; i += 256) x[i] = xs[i];
}

__global__ void bwd_solve_vec_k(const float* __restrict__ L, const float* __restrict__ b,
                                float* __restrict__ x) {
  __shared__ float xs[N];
  __shared__ float red[256];
  int tid = threadIdx.x;
  for (int i = tid; i < N; i += 256) xs[i] = b[i];
  __syncthreads();
  for (int i = N - 1; i >= 0; --i) {
    float part = 0.0f;
    for (int j = i + 1 + tid; j < N; j += 256) part += L[(size_t)j * N + i] * xs[j];
    red[tid] = part;
    __syncthreads();
    for (int o = 128; o > 0; o >>= 1) {
      if (tid < o) red[tid] += red[tid + o];
      __syncthreads();
    }
    if (tid == 0) xs[i] = (xs[i] - red[0]) / L[(size_t)i * N + i];
    __syncthreads();
  }
  for (int i = tid; i < N; i += 256) x[i] = xs[i];
}

// ---------------- sampling kernels ----------------
__global__ void normal_fill_k(float* __restrict__ out, int count, unsigned seed) {
  for (int i = blockIdx.x * blockDim.x + threadIdx.x; i < count;
       i += gridDim.x * blockDim.x)
    out[i] = normal_from(seed + (unsigned)i * 0x9e3779b9u);
}

__global__ void add_normal_k(float* __restrict__ v, unsigned seed) {
  int i = blockIdx.x * blockDim.x + threadIdx.x;
  if (i < N) v[i] += normal_from(seed + (unsigned)i * 0x85ebca6bu);
}

// initial f[d] = L @ eps0[d]  (L lower triangular)
__global__ void gemv_lower_k(const float* __restrict__ L, const float* __restrict__ eps,
                             float* __restrict__ f) {
  int t = blockIdx.x * blockDim.x + threadIdx.x;
  if (t >= ND * N) return;
  int d = t / N, i = t % N;
  const float* row = L + (size_t)i * N;
  const float* e = eps + (size_t)d * N;
  float s = 0.0f;
  for (int j = 0; j <= i; ++j) s += row[j] * e[j];
  f[t] = s;
}

// PG(1, c) via truncated gamma convolution (G terms)
__global__ void pg_draw_k(const float* __restrict__ f, float* __restrict__ omega,
                          unsigned seed) {
  int t = blockIdx.x * blockDim.x + threadIdx.x;
  if (t >= ND * N) return;
  float c = f[t];
  float c2 = c * c * (1.0f / (4.0f * PI_F * PI_F));
  unsigned base = seed * 2654435761u + (unsigned)t * 747796405u;
  float s = 0.0f;
  for (int k = 1; k <= G; ++k) {
    float u = uniform01(hash_u32(base + (unsigned)k * 2891336453u));
    float e = -logf(u);
    float km = (float)k - 0.5f;
    s += e / (km * km + c2);
  }
  omega[t] = s * (1.0f / (2.0f * PI_F * PI_F));
}

// ---------------- final likelihood pieces ----------------
__global__ void z_init_k(float* __restrict__ z, const float* __restrict__ kappa,
                         const float* __restrict__ om) {
  int i = blockIdx.x * blockDim.x + threadIdx.x;
  if (i < N) {
    float o = om[i]; o = o > 1e-16f ? o : 1e-16f;
    z[i] = kappa[i] / o;
  }
}

__global__ void final_accum_k(const float* __restrict__ sol, const float* __restrict__ Lf,
                              const float* __restrict__ om, const float* __restrict__ kappa,
                              float* __restrict__ accum, int d) {
  __shared__ float r0[256], r1[256], r2[256], r3[256];
  int tid = threadIdx.x;
  float quad = 0.0f, logdet = 0.0f, slo = 0.0f, sko = 0.0f;
  for (int i = tid; i < N; i += 256) {
    float s = sol[i];
    quad += s * s;
    logdet += logf(Lf[(size_t)i * N + i]);
    float o = om[i]; o = o > 1e-16f ? o : 1e-16f;
    slo += logf(o);
    sko += kappa[i] * kappa[i] / o;
  }
  r0[tid] = quad; r1[tid] = logdet; r2[tid] = slo; r3[tid] = sko;
  __syncthreads();
  for (int o = 128; o > 0; o >>= 1) {
    if (tid < o) {
      r0[tid] += r0[tid + o]; r1[tid] += r1[tid + o];
      r2[tid] += r2[tid + o]; r3[tid] += r3[tid + o];
    }
    __syncthreads();
  }
  if (tid == 0)
    accum[d] = -0.5f * r0[0] - r1[0] - 0.5f * r2[0] + 0.5f * r3[0]
               - (float)N * 0.69314718056f;
}

__global__ void finalize_k(const float* __restrict__ accum, float* __restrict__ out) {
  if (threadIdx.x == 0) {
    float s = 0.0f;
    for (int d = 0; d < ND; ++d) s += accum[d];
    float mll = s / (float)ND;
    out[0] = -mll / (float)N;
  }
}

// ---------------- host-side orchestration ----------------
static void run_chol(float* M, hipStream_t stream) {
  for (int jb = 0; jb < NB; ++jb) {
    chol_diag_k<<<1, 32, 0, stream>>>(M, jb);
    int rows = N - (jb + 1) * 16;
    if (rows > 0) {
      chol_panel_k<<<(rows + 127) / 128, 128, 0, stream>>>(M, jb);
      int T = NB - (jb + 1);
      chol_syrk_k<<<dim3(T, T), 32, 0, stream>>>(M, jb);
    }
  }
}

extern "C" void kernel_launch(void* const* d_in, const int* in_sizes, int n_in,
                              void* d_out, int out_size, void* d_ws, size_t ws_size,
                              hipStream_t stream) {
  (void)in_sizes; (void)n_in; (void)out_size; (void)ws_size;
  const float* X   = (const float*)d_in[0];
  const int*   Y   = (const int*)d_in[1];
  const float* osp = (const float*)d_in[2];
  const float* lsp = (const float*)d_in[3];
  float* out = (float*)d_out;

  float* ws    = (float*)d_ws;
  float* K     = ws;                 // N*N
  float* Lf    = ws + (size_t)N * N; // N*N working factor
  float* Kinv  = ws + (size_t)2 * N * N; // N*N (also TRSM RHS)
  float* x2    = ws + (size_t)3 * N * N; // N
  float* kappa = x2 + N;             // N
  float* tv    = kappa + N;          // N
  float* tv2   = tv + N;             // N
  float* f     = tv2 + N;            // ND*N
  float* omega = f + ND * N;         // ND*N
  float* eps0  = omega + ND * N;     // ND*N
  float* accum = eps0 + ND * N;      // ND

  // K matrix + kappa
  row_norms_k<<<(N + 255) / 256, 256, 0, stream>>>(X, x2);
  rbf_wmma_k<<<dim3(NB, NB), 32, 0, stream>>>(X, x2, osp, lsp, K);
  kappa_init_k<<<(N + 255) / 256, 256, 0, stream>>>(Y, kappa);

  // L = chol(K + jitter I)
  form_matrix_k<<<256, 256, 0, stream>>>(Lf, K, nullptr, JITTER, 0);
  run_chol(Lf, stream);

  // Kinv = L^-T L^-1 via two blocked TRSMs against identity
  set_identity_k<<<256, 256, 0, stream>>>(Kinv);
  for (int jb = 0; jb < NB; ++jb) {
    if (jb > 0) trsm_fwd_gemm_k<<<NB, 32, 0, stream>>>(Kinv, Lf, jb);
    trsm_fwd_small_k<<<4, 256, 0, stream>>>(Kinv, Lf, jb);
  }
  for (int jb = NB - 1; jb >= 0; --jb) {
    if (jb < NB - 1) trsm_bwd_gemm_k<<<NB, 32, 0, stream>>>(Kinv, Lf, jb);
    trsm_bwd_small_k<<<4, 256, 0, stream>>>(Kinv, Lf, jb);
  }

  // initial state: f = L @ eps0, omega = PG(f)
  normal_fill_k<<<(ND * N + 255) / 256, 256, 0, stream>>>(eps0, ND * N, 1u);
  gemv_lower_k<<<(ND * N + 255) / 256, 256, 0, stream>>>(Lf, eps0, f);
  pg_draw_k<<<(ND * N + 255) / 256, 256, 0, stream>>>(f, omega, 100u);

  // Gibbs loop: f_d = La^{-T}(La^{-1} kappa + eps), A = Kinv + diag(omega_d)
  for (int s = 0; s < STEPS; ++s) {
    for (int d = 0; d < ND; ++d) {
      form_matrix_k<<<256, 256, 0, stream>>>(Lf, Kinv, omega + (size_t)d * N, 0.0f, 1);
      run_chol(Lf, stream);
      fwd_solve_vec_k<<<1, 256, 0, stream>>>(Lf, kappa, tv);
      add_normal_k<<<4, 256, 0, stream>>>(tv, 1000u + (unsigned)(s * ND + d));
      bwd_solve_vec_k<<<1, 256, 0, stream>>>(Lf, tv, f + (size_t)d * N);
    }
    pg_draw_k<<<(ND * N + 255) / 256, 256, 0, stream>>>(f, omega, 200u + (unsigned)s);
  }

  // final marginal likelihood: chol(K + diag(1/om_d)), solve, reduce
  for (int d = 0; d < ND; ++d) {
    form_matrix_k<<<256, 256, 0, stream>>>(Lf, K, omega + (size_t)d * N, 0.0f, 2);
    run_chol(Lf, stream);
    z_init_k<<<4, 256, 0, stream>>>(tv, kappa, omega + (size_t)d * N);
    fwd_solve_vec_k<<<1, 256, 0, stream>>>(Lf, tv, tv2);
    final_accum_k<<<1, 256, 0, stream>>>(tv2, Lf, omega + (size_t)d * N, kappa, accum, d);
  }
  finalize_k<<<1, 32, 0, stream>>>(accum, out);
}